// FtxScaleLayer_9096740733430
// MI455X (gfx1250) — compile-verified
//
#include <hip/hip_runtime.h>

// CDNA5 / gfx1250: per-feature scale out = x * pars implemented through the
// f32 WMMA pipe as x @ diag(pars), tiled 16x16, D = A(16x4) x B(4x16) + C
// with B a diagonal chunk built in registers. Bit-exact vs scalar multiply
// (one nonzero product per accumulation chain, RNE). Bandwidth-bound:
// ~537 MB @ 23.3 TB/s => ~23 us floor; WMMA issue cost is negligible.

typedef float v2f __attribute__((ext_vector_type(2)));
typedef float v8f __attribute__((ext_vector_type(8)));

#define FEATURES 256
#define TILE 16

__global__ __launch_bounds__(256) void ftx_scale_wmma_kernel(
    const float* __restrict__ x,
    const float* __restrict__ pars,
    float* __restrict__ out,
    int nrows)
{
    const int lane  = threadIdx.x & 31;          // wave32
    const int waveB = threadIdx.x >> 5;          // wave index in block (0..7)
    const int wave  = blockIdx.x * (blockDim.x >> 5) + waveB;
    const int m0    = wave * TILE;               // first row of this wave's 16-row panel
    if (m0 >= nrows) return;                     // wave-uniform guard (EXEC stays all-ones)

    const int n  = lane & 15;                    // N-column (and A-row M) id per ISA layout
    const int hi = lane >> 4;                    // half-wave: selects K=0..1 vs K=2..3

    // A loads: lane holds row m0+n of x. D stores: lane writes column c+n of
    // rows m0 + v + 8*hi (C/D VGPR layout: VGPRv -> M=v (lanes 0-15), M=v+8 (16-31)).
    const float* __restrict__ xrow  = x   + (size_t)(m0 + n) * FEATURES;
    float*       __restrict__ obase = out + (size_t)(m0 + 8 * hi) * FEATURES + n;

    #pragma unroll 1
    for (int t = 0; t < FEATURES / TILE; ++t) {
        const int c = t * TILE;
        const float p = pars[c + n];             // diagonal value for column c+n (hot, L0)

        // Prefetch next tile's A data (global_prefetch_b8 on gfx1250).
        if (t + 1 < FEATURES / TILE)
            __builtin_prefetch(xrow + c + TILE + 2 * hi, 0, 1);

        v8f acc = {};
        #pragma unroll
        for (int j = 0; j < 4; ++j) {
            // 16x4 f32 A layout: VGPR0 = K=k0, VGPR1 = K=k0+1, k0 = 2*hi (+4j tile step)
            const int k0 = 4 * j + 2 * hi;
            v2f a = *(const v2f*)(xrow + c + k0);

            // B(4x16) diagonal chunk: B[k, nn] = pars[c+nn] iff nn == 4j + k.
            // Layout mirrors A: VGPR0 holds rows K=2*hi, VGPR1 rows K=2*hi+1,
            // N striped across lanes 0-15 / 16-31.
            v2f b;
            b.x = (n == k0)     ? p : 0.0f;
            b.y = (n == k0 + 1) ? p : 0.0f;

            acc = __builtin_amdgcn_wmma_f32_16x16x4_f32(
                /*neg_a=*/false, a, /*neg_b=*/false, b,
                /*c_mod=*/(short)0, acc, /*reuse_a=*/false, /*reuse_b=*/false);
        }

        // Store D tile: per VGPR v, lanes 0-15 write 64B of row m0+v, lanes
        // 16-31 write 64B of row m0+v+8 — fully coalesced. Output is streamed
        // once (256 MB), so use non-temporal stores.
        #pragma unroll
        for (int v = 0; v < 8; ++v) {
            __builtin_nontemporal_store(acc[v], obase + (size_t)v * FEATURES + c);
        }
    }
}

extern "C" void kernel_launch(void* const* d_in, const int* in_sizes, int n_in,
                              void* d_out, int out_size, void* d_ws, size_t ws_size,
                              hipStream_t stream) {
    const float* x    = (const float*)d_in[0];
    const float* pars = (const float*)d_in[1];
    float*       out  = (float*)d_out;

    const int features = in_sizes[1];            // 256
    const int nrows    = in_sizes[0] / features; // 262144
    (void)features;

    const int wavesNeeded = nrows / TILE;        // one wave per 16-row panel
    const int blocks      = (wavesNeeded + 7) / 8;  // 8 waves (256 thr) per block

    ftx_scale_wmma_kernel<<<blocks, 256, 0, stream>>>(x, pars, out, nrows);
}